// PolyGCN_34316788695555
// MI455X (gfx1250) — compile-verified
//
#include <hip/hip_runtime.h>

typedef __attribute__((ext_vector_type(2))) float v2f;
typedef __attribute__((ext_vector_type(8))) float v8f;

// ---------------------------------------------------------------------------
// Kernel 1: T[k] = X @ W[k]        X:[N,DIN] row-major, W:[K,DIN,DO], T:[K,N,DO]
// One wave = one 16-row block for one k (blockIdx.y). Reduction DIN via
// f32 WMMA 16x16x4. DIN=64 -> 16 steps, fully unrolled.
// ---------------------------------------------------------------------------
template <int DIN, int DO>
__global__ __launch_bounds__(256) void polygcn_gemm_xw(
    const float* __restrict__ X, const float* __restrict__ W,
    float* __restrict__ T, int N, int nRB) {
  const int lane = threadIdx.x & 31;
  const int wv   = threadIdx.x >> 5;
  const int rb   = blockIdx.x * 8 + wv;
  if (rb >= nRB) return;                       // wave-uniform exit
  const int half = lane >> 4;                  // 0 or 1
  const int lr   = lane & 15;
  const int r0   = rb * 16;

  const float* Wk = W + (size_t)blockIdx.y * DIN * DO;
  float*       Tk = T + (size_t)blockIdx.y * N * DO;

  v8f acc[DO / 16] = {};

  const float* arow = X + (size_t)(r0 + lr) * DIN + 2 * half;
#pragma unroll
  for (int m = 0; m < DIN; m += 4) {
    v2f a;
    a.x = arow[m];
    a.y = arow[m + 1];
    const float* bp = Wk + (size_t)(m + 2 * half) * DO + lr;
#pragma unroll
    for (int ct = 0; ct < DO / 16; ++ct) {
      v2f b;
      b.x = bp[ct * 16];
      b.y = bp[DO + ct * 16];
      acc[ct] = __builtin_amdgcn_wmma_f32_16x16x4_f32(
          false, a, false, b, (short)0, acc[ct], false, false);
    }
  }

  float* orow = Tk + (size_t)(r0 + 8 * half) * DO + lr;
#pragma unroll
  for (int ct = 0; ct < DO / 16; ++ct) {
#pragma unroll
    for (int j = 0; j < 8; ++j) {
      orow[(size_t)j * DO + ct * 16] = acc[ct][j];
    }
  }
}

// ---------------------------------------------------------------------------
// Kernel 2: PART[k] = poly[k] @ T[k]    poly:[K,N,N], T:[K,N,DO], PART:[K,N,DO]
// One wave = one 16-row output block, all DO columns, for k = blockIdx.y.
// B operand (rows of T[k]) is identical for the 8 waves of a workgroup, so it
// is staged into a double-buffered LDS tile with GLOBAL_LOAD_ASYNC_TO_LDS_B128
// (ASYNCcnt path): the next chunk's copy is issued before computing the
// current chunk, fully overlapping with the WMMA burst + poly HBM stream.
// Chunk = 80 reduction rows (10000 = 125 * 80).
// ---------------------------------------------------------------------------
template <int DO>
__global__ __launch_bounds__(256) void polygcn_poly_agg(
    const float* __restrict__ P, const float* __restrict__ T,
    float* __restrict__ PART, int N, int nRB) {
  constexpr int MCH = 80;                      // reduction-chunk rows in LDS
  constexpr int NV  = (MCH * DO) / 4;          // float4 elements per chunk
  __shared__ float ldsB[2][MCH * DO];

  const int lane = threadIdx.x & 31;
  const int wv   = threadIdx.x >> 5;
  const int rb   = blockIdx.x * 8 + wv;
  const bool active = (rb < nRB);              // wave-uniform
  const int half = lane >> 4;
  const int lr   = lane & 15;
  const int r0   = rb * 16;

  const float* Pk = P + (size_t)blockIdx.y * N * N;
  const float* Tk = T + (size_t)blockIdx.y * N * DO;
  float*    PARTk = PART + (size_t)blockIdx.y * N * DO;

  const float* arow =
      active ? (Pk + (size_t)(r0 + lr) * N + 2 * half) : Pk;

  // Low 32 bits of the generic shared-memory address == LDS byte offset.
  const unsigned lds0 = (unsigned)(size_t)&ldsB[0][0];
  const unsigned lds1 = (unsigned)(size_t)&ldsB[1][0];

  // Issue async copy of one 80-row chunk of T[k] into an LDS buffer.
  auto stage = [&](int chunk, unsigned ldsbase) {
    const unsigned long long sbase =
        (unsigned long long)(size_t)(Tk + (size_t)chunk * MCH * DO);
    for (int i = threadIdx.x; i < NV; i += 256) {
      const unsigned ldsa = ldsbase + (unsigned)i * 16u;
      const unsigned voff = (unsigned)i * 16u;
      asm volatile("global_load_async_to_lds_b128 %0, %1, %2"
                   :: "v"(ldsa), "v"(voff), "s"(sbase)
                   : "memory");
    }
  };

  const int nChunks = N / MCH;
  v8f acc[DO / 16] = {};

  // Prologue: stage chunk 0, wait, make visible to all waves.
  stage(0, lds0);
  asm volatile("s_wait_asynccnt 0x0" ::: "memory");
  __syncthreads();

  for (int c = 0; c < nChunks; ++c) {
    const int curBuf = (c & 1);
    if (c + 1 < nChunks) stage(c + 1, curBuf ? lds0 : lds1);

    if (active) {
      // prefetch this wave's poly rows one chunk ahead
      __builtin_prefetch(arow + (c + 1) * MCH, 0, 0);
      const float* bbase = &ldsB[curBuf][0];
      const int mb = c * MCH;
#pragma unroll
      for (int mi = 0; mi < MCH; mi += 4) {
        v2f a;
        a.x = arow[mb + mi];
        a.y = arow[mb + mi + 1];
        const float* bp = bbase + (mi + 2 * half) * DO + lr;
#pragma unroll
        for (int ct = 0; ct < DO / 16; ++ct) {
          v2f b;
          b.x = bp[ct * 16];
          b.y = bp[DO + ct * 16];
          acc[ct] = __builtin_amdgcn_wmma_f32_16x16x4_f32(
              false, a, false, b, (short)0, acc[ct], false, false);
        }
      }
    }
    // Wait for this wave's async issues, then workgroup barrier: next chunk's
    // data visible to all waves; all readers done before buffer reuse.
    asm volatile("s_wait_asynccnt 0x0" ::: "memory");
    __syncthreads();
  }

  if (active) {
    float* orow = PARTk + (size_t)(r0 + 8 * half) * DO + lr;
#pragma unroll
    for (int ct = 0; ct < DO / 16; ++ct) {
#pragma unroll
      for (int j = 0; j < 8; ++j) {
        orow[(size_t)j * DO + ct * 16] = acc[ct][j];
      }
    }
  }
}

// ---------------------------------------------------------------------------
// Kernel 3: out[i] = (relu?) sum_k part[k*total + i]   (deterministic reduce)
// ---------------------------------------------------------------------------
template <bool RELU>
__global__ __launch_bounds__(256) void polygcn_reduce(
    const float* __restrict__ part, float* __restrict__ out, int total, int K) {
  const int i = blockIdx.x * blockDim.x + threadIdx.x;
  if (i >= total) return;
  float v = 0.0f;
  for (int k = 0; k < K; ++k) v += part[(size_t)k * total + i];
  if (RELU) v = fmaxf(v, 0.0f);
  out[i] = v;
}

// ---------------------------------------------------------------------------
// Launch:  t1 = x@W1 ; p1[k] = poly[k]@t1[k] ; h = relu(sum p1)
//          t2 = h@W2 ; p2[k] = poly[k]@t2[k] ; out = sum p2
// ---------------------------------------------------------------------------
extern "C" void kernel_launch(void* const* d_in, const int* in_sizes, int n_in,
                              void* d_out, int out_size, void* d_ws,
                              size_t ws_size, hipStream_t stream) {
  const float* x    = (const float*)d_in[0];   // [N, 64]
  const float* poly = (const float*)d_in[1];   // [K, N, N]
  const float* W1   = (const float*)d_in[2];   // [K, 64, 64]
  const float* W2   = (const float*)d_in[3];   // [K, 64, 32]
  float* out = (float*)d_out;                  // [N, 32]

  const int DIN = 64, DH = 64, DOUT = 32;
  const int N = in_sizes[0] / DIN;             // 10000
  const int K = in_sizes[2] / (DIN * DH);      // 3
  const int nRB = N / 16;                      // 625
  const int blocks = (nRB + 7) / 8;            // 79

  // workspace layout (floats)
  float* t1 = (float*)d_ws;                    // K*N*DH
  float* p1 = t1 + (size_t)K * N * DH;         // K*N*DH
  float* h  = p1 + (size_t)K * N * DH;         // N*DH
  float* t2 = h  + (size_t)N * DH;             // K*N*DOUT
  float* p2 = t2 + (size_t)K * N * DOUT;       // K*N*DOUT

  // Layer 1
  polygcn_gemm_xw<64, 64><<<dim3(blocks, K), 256, 0, stream>>>(x, W1, t1, N, nRB);
  polygcn_poly_agg<64><<<dim3(blocks, K), 256, 0, stream>>>(poly, t1, p1, N, nRB);
  {
    const int total = N * DH;
    polygcn_reduce<true><<<(total + 255) / 256, 256, 0, stream>>>(p1, h, total, K);
  }

  // Layer 2
  polygcn_gemm_xw<64, 32><<<dim3(blocks, K), 256, 0, stream>>>(h, W2, t2, N, nRB);
  polygcn_poly_agg<32><<<dim3(blocks, K), 256, 0, stream>>>(poly, t2, p2, N, nRB);
  {
    const int total = N * DOUT;
    polygcn_reduce<false><<<(total + 255) / 256, 256, 0, stream>>>(p2, out, total, K);
  }
}